// TorchModalitySampler_31224412242713
// MI455X (gfx1250) — compile-verified
//
#include <hip/hip_runtime.h>
#include <stdint.h>

// TorchModalitySampler for MI455X (gfx1250, wave32).
// One workgroup (16 wave32) per 512x512 image. Single async-pipelined read
// pass builds a per-column argmax cache; 5 incremental 9-column repairs
// handle the iterative peak suppression. Input is never mutated
// (suppression is applied as a register-side mask of zeroed rects).

#define HH   512
#define WW   512
#define REC  5
#define RAD  2
#define PW   508          // 512 - 5 + 1
#define NT   6
#define INVM 262143       // 2^18-1 >= max flat index (507*508+507 = 258063)

typedef unsigned long long u64;

#if defined(__has_builtin)
#  if __has_builtin(__builtin_amdgcn_global_load_async_to_lds_b128)
#    define USE_ASYNC_LDS 1
#  endif
#  if __has_builtin(__builtin_amdgcn_s_wait_asynccnt)
#    define USE_WAIT_BUILTIN 1
#  endif
#endif
#ifndef USE_ASYNC_LDS
#  define USE_ASYNC_LDS 0
#endif
#ifndef USE_WAIT_BUILTIN
#  define USE_WAIT_BUILTIN 0
#endif

// Pointer types matching the builtin signature: (v4i AS1*, v4i AS3*, Ii, Ii).
typedef int v4i_t __attribute__((vector_size(16)));
typedef __attribute__((address_space(1))) v4i_t* gv4i_p;
typedef __attribute__((address_space(3))) v4i_t* lv4i_p;

__device__ __forceinline__ void async_row16(const float* gsrc, float* lds_dst) {
#if USE_ASYNC_LDS
  __builtin_amdgcn_global_load_async_to_lds_b128(
      (gv4i_p)(uintptr_t)gsrc,
      (lv4i_p)(uint32_t)(uintptr_t)lds_dst,
      0, 0);
#else
  *(float4*)lds_dst = *(const float4*)gsrc;
#endif
}

#if USE_ASYNC_LDS
#  if USE_WAIT_BUILTIN
#    define WAIT_ASYNC(N) do { __builtin_amdgcn_s_wait_asynccnt(N); \
                               asm volatile("" ::: "memory"); } while (0)
#  else
#    define WAIT_ASYNC(N) asm volatile("s_wait_asynccnt " #N ::: "memory")
#  endif
#else
#  define WAIT_ASYNC(N) asm volatile("" ::: "memory")
#endif

__global__ __launch_bounds__(512)
void modality_sampler_kernel(const float* __restrict__ hm_all, int* __restrict__ out) {
  const int img  = blockIdx.x;
  const float* hm = hm_all + (size_t)img * HH * WW;
  const int tid  = threadIdx.x;        // == column index c for the main pass
  const int lane = tid & 31;
  const int wid  = tid >> 5;           // 16 waves

  // Per-wave private row ring: wave w stages columns [32w, 32w+35] (32 + 4 halo).
  __shared__ alignas(16) float rowbuf[16][8][36];   // 18.4 KB
  __shared__ u64 colbest[PW];                       // per-column packed argmax
  __shared__ u64 redbuf[16];
  __shared__ u64 gbest;
  __shared__ int rect_r[NT], rect_c[NT];

  // Hoisted per-lane async source/dest bases (one B128 per wave per row,
  // lanes 0..8 each move 4 floats; last wave's halo clamped to col 508).
  const float* gsrc0;
  float* ldst0;
  {
    int gc = wid * 32 + 4 * lane;
    if (gc > WW - 4) gc = WW - 4;
    gsrc0 = hm + gc;
    ldst0 = &rowbuf[wid][0][4 * lane];
  }
  auto issue_row = [&](int y) {
    if (lane < 9) async_row16(gsrc0 + (size_t)y * WW, ldst0 + (y & 7) * 36);
  };
  const float* rb0 = &rowbuf[wid][0][lane];         // hoisted read base

  // ---------------- Pass 1: streaming box-sum + per-column argmax ----------
  // vs = vertical running sum of horizontal 5-sums; 5-deep register ring with
  // static indexing (unrolled x5, y%5 == k). No workgroup barriers in the
  // sweep: each wave waits only on its own ASYNCcnt (prefetch depth 2,
  // LDS ring depth 8). Tracking is float best / int row (strict-greater ==
  // first occurrence); the packed u64 key is built once per column at the end.
  float ring[5];
  float vs = 0.f;

  issue_row(0);
  issue_row(1);

  // Prologue rows 0..4: fill pipeline + ring; row 4 yields agg row 0.
#pragma unroll
  for (int k = 0; k < 5; ++k) {                     // y == k
    issue_row(k + 2);
    WAIT_ASYNC(2);
    const float* rb = rb0 + (k & 7) * 36;
    float h = rb[0] + rb[1] + rb[2] + rb[3] + rb[4];
    vs += h;
    ring[k] = h;
  }
  float best_v = vs;                                // agg row 0 (y = 4)
  int   best_y = 4;

  // Main loop rows 5..509 (505 = 5*101): fully uniform, constant waits.
  for (int yb = 5; yb < 510; yb += 5) {
#pragma unroll
    for (int k = 0; k < 5; ++k) {
      const int y = yb + k;                         // y % 5 == k
      issue_row(y + 2);
      WAIT_ASYNC(2);
      const float* rb = rb0 + (y & 7) * 36;
      float h = rb[0] + rb[1] + rb[2] + rb[3] + rb[4];
      vs += h - ring[k];                            // ring[k] holds h from y-5
      ring[k] = h;
      if (vs > best_v) { best_v = vs; best_y = y; }
    }
  }

  // Epilogue rows 510 (k=0) and 511 (k=1): drain the async pipeline.
  {
    WAIT_ASYNC(1);
    const float* rb = rb0 + (510 & 7) * 36;
    float h = rb[0] + rb[1] + rb[2] + rb[3] + rb[4];
    vs += h - ring[0];
    ring[0] = h;
    if (vs > best_v) { best_v = vs; best_y = 510; }

    WAIT_ASYNC(0);
    rb = rb0 + (511 & 7) * 36;
    h = rb[0] + rb[1] + rb[2] + rb[3] + rb[4];
    vs += h - ring[1];
    if (vs > best_v) { best_v = vs; best_y = 511; }
  }

  if (tid < PW) {
    int flat = (best_y - (REC - 1)) * PW + tid;
    colbest[tid] = ((u64)__float_as_uint(best_v) << 18) | (u64)(INVM - flat);
  }
  __syncthreads();

  // ---------------- 6 targets: global reduce + incremental repair ----------
  for (int t = 0; t < NT; ++t) {
    // Global argmax over per-column bests (wave32 shuffle tree + 16-slot LDS).
    u64 key = (tid < PW) ? colbest[tid] : 0ull;
#pragma unroll
    for (int m = 16; m >= 1; m >>= 1) {
      u64 o = __shfl_xor(key, m, 32);
      key = key > o ? key : o;
    }
    if (lane == 0) redbuf[wid] = key;
    __syncthreads();
    if (tid == 0) {
      u64 m = 0;
#pragma unroll
      for (int wq = 0; wq < 16; ++wq) m = m > redbuf[wq] ? m : redbuf[wq];
      gbest = m;
    }
    __syncthreads();
    u64 g = gbest;
    int flat = INVM - (int)(g & (u64)INVM);
    int r = flat / PW;
    int c = flat - r * PW;
    if (tid == 0) {
      out[img * (NT * 2) + 2 * t + 0] = c + RAD;    // SWAP_RC: [col, row]
      out[img * (NT * 2) + 2 * t + 1] = r + RAD;
      rect_r[t] = r;
      rect_c[t] = c;
    }
    __syncthreads();
    if (t == NT - 1) break;

    // Suppression of hm[r:r+5, c:c+5] only changes agg columns c-4..c+4.
    // Recompute those 9 columns over all rows with rect masking (input stays
    // untouched). Thread i == agg row; needs hm[i..i+4][c-4..c+8] (<=65 loads).
    int rr[NT], rc[NT];
#pragma unroll
    for (int q = 0; q < NT; ++q) {
      rr[q] = rect_r[q];
      rc[q] = rect_c[q];
    }

    float colsum[13];
#pragma unroll
    for (int k = 0; k < 13; ++k) colsum[k] = 0.f;
    if (tid < PW) {
#pragma unroll
      for (int dy = 0; dy < REC; ++dy) {
        int y = tid + dy;
        const float* rowp = hm + (size_t)y * WW;
#pragma unroll
        for (int k = 0; k < 13; ++k) {
          int x = c - 4 + k;
          float v = 0.f;
          if (x >= 0 && x < WW) v = rowp[x];
#pragma unroll
          for (int q = 0; q < NT; ++q) {
            if (q <= t && (unsigned)(y - rr[q]) < REC && (unsigned)(x - rc[q]) < REC)
              v = 0.f;
          }
          colsum[k] += v;
        }
      }
    }

#pragma unroll
    for (int jj = 0; jj < 9; ++jj) {                // fully unrolled: static colsum idx
      int j = c - 4 + jj;
      u64 k2 = 0;
      if (tid < PW && j >= 0 && j < PW) {
        float agg = colsum[jj] + colsum[jj + 1] + colsum[jj + 2] +
                    colsum[jj + 3] + colsum[jj + 4];
        int fl = tid * PW + j;
        k2 = ((u64)__float_as_uint(agg) << 18) | (u64)(INVM - fl);
      }
#pragma unroll
      for (int m = 16; m >= 1; m >>= 1) {
        u64 o = __shfl_xor(k2, m, 32);
        k2 = k2 > o ? k2 : o;
      }
      if (lane == 0) redbuf[wid] = k2;
      __syncthreads();
      if (tid == 0) {
        u64 m = 0;
#pragma unroll
        for (int wq = 0; wq < 16; ++wq) m = m > redbuf[wq] ? m : redbuf[wq];
        if (j >= 0 && j < PW) colbest[j] = m;
      }
      __syncthreads();
    }
  }
}

extern "C" void kernel_launch(void* const* d_in, const int* in_sizes, int n_in,
                              void* d_out, int out_size, void* d_ws, size_t ws_size,
                              hipStream_t stream) {
  (void)n_in; (void)out_size; (void)d_ws; (void)ws_size;
  const float* hm = (const float*)d_in[0];
  const int B = in_sizes[0] / (HH * WW);            // 128 images
  modality_sampler_kernel<<<dim3(B), dim3(512), 0, stream>>>(hm, (int*)d_out);
}